// EventWarping_18442589569626
// MI455X (gfx1250) — compile-verified
//
#include <hip/hip_runtime.h>
#include <stdint.h>

// ---------------- problem constants (from reference) ----------------
#define H_ 256
#define W_ 256
#define HW_ (H_ * W_)
#define FLOW_SCALING_ 256.0f
#define EPS_ 1e-9f
// planes: (tref in {1.0 fw, 0.0 bw}) x (batch in {0,1})
#define NPLANES 4
// per pixel, 4 floats: {pos_w, pos_tsw, neg_w, neg_tsw}
#define HIST_FLOATS (NPLANES * HW_ * 4)

// ---------------- reduction tiling ----------------
#define TILE_PX 4096   // pixels per block  (64 KB of float4 in LDS)
#define WAVE_PX 512    // pixels per wave   (8 KB slice -> one TDM descriptor)
#define RED_THREADS 256

// CDNA5 Tensor Data Mover: this toolchain declares the 6-arg builtin
// (uint32x4 g0, int32x8 g1, int32x4 g2, int32x4 g3, int32x8 g4, i32 cpol).
#if defined(__has_builtin)
#if __has_builtin(__builtin_amdgcn_tensor_load_to_lds) && \
    __has_builtin(__builtin_amdgcn_s_wait_tensorcnt)
#define USE_TDM 1
#endif
#endif
#ifndef USE_TDM
#define USE_TDM 0
#endif

typedef unsigned int u32x4 __attribute__((ext_vector_type(4)));
typedef int          i32x8 __attribute__((ext_vector_type(8)));
typedef int          i32x4 __attribute__((ext_vector_type(4)));
typedef float        f32x4 __attribute__((ext_vector_type(4)));
typedef float        f32x2 __attribute__((ext_vector_type(2)));

__device__ __forceinline__ void atomic_add_f32(float* p, float v) {
  // relaxed, device-scope, result unused -> global_atomic_add_f32 (no return)
  (void)__hip_atomic_fetch_add(p, v, __ATOMIC_RELAXED, __HIP_MEMORY_SCOPE_AGENT);
}

// ---------------- kernel: fused fw+bw bilinear splat (critical path) -------
// First in the file so the disasm snippet shows its atomic lowering.
// One thread per event; reads the event once (non-temporal: streamed exactly
// once, keep L2 for the accumulators), splats for both tref=1 and tref=0.
// 16 global_atomic_add_f32 per event into the L2-resident 4 MB accumulator.
__global__ void ew_splat(const float* __restrict__ ev_,
                         const float* __restrict__ fl_,
                         float* __restrict__ hist, int N, int total) {
  int i = blockIdx.x * blockDim.x + threadIdx.x;
  if (i >= total) return;
  f32x4 e = __builtin_nontemporal_load((const f32x4*)ev_ + i); // (ts,x,y,p)
  f32x2 f = __builtin_nontemporal_load((const f32x2*)fl_ + i);
  float ts = e.x, x = e.y, y = e.z;
  float fx = f.x * FLOW_SCALING_;
  float fy = f.y * FLOW_SCALING_;
  int b = (i >= N) ? 1 : 0;       // B == 2 (reference setup)
  int chanOff = (e.w > 0.0f) ? 0 : 2;

#pragma unroll
  for (int t = 0; t < 2; ++t) {
    float tref = (t == 0) ? 1.0f : 0.0f;
    float dt = tref - ts;
    float wx = x + dt * fx;
    float wy = y + dt * fy;
    float lx = floorf(wx), ly = floorf(wy);
    float ax = wx - lx, ay = wy - ly;   // in [0,1)
    int ix = (int)lx, iy = (int)ly;
    float* plane = hist + (size_t)((t * 2 + b) * HW_ * 4);

    float wgt[4] = {(1.0f - ax) * (1.0f - ay), ax * (1.0f - ay),
                    (1.0f - ax) * ay,          ax * ay};
    int cx[4] = {ix, ix + 1, ix,     ix + 1};
    int cy[4] = {iy, iy,     iy + 1, iy + 1};
#pragma unroll
    for (int c = 0; c < 4; ++c) {
      if ((unsigned)cx[c] < (unsigned)W_ && (unsigned)cy[c] < (unsigned)H_) {
        float* pp = plane + (((cy[c] << 8) + cx[c]) * 4 + chanOff);
        atomic_add_f32(pp,     wgt[c]);        // iwe
        atomic_add_f32(pp + 1, ts * wgt[c]);   // ts-weighted iwe
      }
    }
  }
}

// ---------------- kernel: per-plane reduction (TDM -> LDS -> shuffle) ------
__global__ __launch_bounds__(RED_THREADS) void ew_reduce(
    const float* __restrict__ hist, float* __restrict__ partial) {
  const int plane = blockIdx.y;
  const int chunk = blockIdx.x;
  const int lane  = threadIdx.x & 31;
  const int wave  = threadIdx.x >> 5;
  const float* src =
      hist + ((size_t)plane * HW_ + (size_t)chunk * TILE_PX) * 4;

  float lsum = 0.0f, nzsum = 0.0f;

#if USE_TDM
  __shared__ float4 tile[TILE_PX];
  {
    // Each wave DMAs its own 8 KB slice (2048 f32) via the Tensor Data Mover.
    const float* gsrc = src + (size_t)wave * WAVE_PX * 4;
    uint64_t ga = (uint64_t)(uintptr_t)gsrc;
    uint32_t lo = (uint32_t)(uintptr_t)(&tile[wave * WAVE_PX]); // LDS byte off
    const unsigned elems = WAVE_PX * 4;  // 2048 elements, data_size = 4B

    // D# group 0 (ISA 8.3): count=1 | lds_addr | global_addr | type=2
    u32x4 g0;
    g0[0] = 1u;
    g0[1] = lo;
    g0[2] = (uint32_t)ga;
    g0[3] = ((uint32_t)(ga >> 32) & 0x01FFFFFFu) | 0x80000000u; // type=2

    // D# group 1 (ISA 8.4): mask=0, data_size=4B, 1-D tile of `elems`
    i32x8 g1;
    g1[0] = (int)(2u << 16);                    // data_size = 2 (4 bytes)
    g1[1] = (int)((elems & 0xFFFFu) << 16);     // tensor_dim0 [15:0]
    g1[2] = (int)((elems >> 16) | (1u << 16));  // tensor_dim0 hi, tensor_dim1=1
    g1[3] = (int)((elems & 0xFFFFu) << 16);     // tile_dim0 = elems
    g1[4] = 0;                                  // tile_dim1 = tile_dim2 = 0
    g1[5] = (int)elems;                         // tensor_dim0_stride lo
    g1[6] = 0;
    g1[7] = 0;
    i32x4 g2 = {0, 0, 0, 0};                    // tensor_dim2.. unused
    i32x4 g3 = {0, 0, 0, 0};                    // tensor_dim3.. unused
    i32x8 g4 = {0, 0, 0, 0, 0, 0, 0, 0};        // unused trailing group

    __builtin_amdgcn_tensor_load_to_lds(g0, g1, g2, g3, g4, 0); // TENSORcnt++
    __builtin_amdgcn_s_wait_tensorcnt(0);                       // DMA done
    __asm__ volatile("" ::: "memory");          // no hoist of LDS reads
  }
#pragma unroll
  for (int j = 0; j < WAVE_PX / 32; ++j) {
    float4 v = tile[wave * WAVE_PX + j * 32 + lane];        // ds_load_b128
    float a = v.y / (v.x + EPS_);
    float b = v.w / (v.z + EPS_);
    lsum += a * a + b * b;
    nzsum += ((v.x + v.z) > 0.0f) ? 1.0f : 0.0f;
  }
#else
  const float4* src4 = (const float4*)src;
#pragma unroll
  for (int j = 0; j < WAVE_PX / 32; ++j) {
    float4 v = src4[wave * WAVE_PX + j * 32 + lane];
    float a = v.y / (v.x + EPS_);
    float b = v.w / (v.z + EPS_);
    lsum += a * a + b * b;
    nzsum += ((v.x + v.z) > 0.0f) ? 1.0f : 0.0f;
  }
#endif

  // wave32 shuffle reduction, wave leaders accumulate globally
  for (int off = 16; off > 0; off >>= 1) {
    lsum  += __shfl_down(lsum,  off, 32);
    nzsum += __shfl_down(nzsum, off, 32);
  }
  if (lane == 0) {
    atomic_add_f32(&partial[plane], lsum);
    atomic_add_f32(&partial[NPLANES + plane], nzsum);
  }
}

// ---------------- kernel: zero workspace (b128 stores) ----------------
__global__ void ew_zero4(f32x4* __restrict__ ws4, int n4) {
  int i = blockIdx.x * blockDim.x + threadIdx.x;
  if (i < n4) {
    f32x4 z = {0.0f, 0.0f, 0.0f, 0.0f};
    ws4[i] = z;
  }
}

// ---------------- kernel: finalize scalar ----------------
__global__ void ew_finalize(const float* __restrict__ partial,
                            float* __restrict__ out) {
  float s = 0.0f;
  for (int p = 0; p < NPLANES; ++p) s += partial[p] / partial[NPLANES + p];
  out[0] = s;  // (fw + bw) / MAX_TS^2, MAX_TS = 1
}

// ---------------- host launcher ----------------
extern "C" void kernel_launch(void* const* d_in, const int* in_sizes, int n_in,
                              void* d_out, int out_size, void* d_ws,
                              size_t ws_size, hipStream_t stream) {
  const float* ev = (const float*)d_in[0];     // [B,N,4] f32
  const float* fl = (const float*)d_in[1];     // [B,N,2] f32
  float* ws = (float*)d_ws;
  float* hist = ws;                            // 4 MiB accumulators
  float* partial = ws + HIST_FLOATS;           // 8 floats: loss[4], nz[4]
  float* out = (float*)d_out;

  int total = in_sizes[0] / 4;                 // B*N events
  int N = total / 2;                           // B == 2 per reference setup

  int n4 = (HIST_FLOATS + 2 * NPLANES) / 4;    // divisible by 4
  ew_zero4<<<(n4 + 255) / 256, 256, 0, stream>>>((f32x4*)ws, n4);
  ew_splat<<<(total + 255) / 256, 256, 0, stream>>>(ev, fl, hist, N, total);
  ew_reduce<<<dim3(HW_ / TILE_PX, NPLANES), RED_THREADS, 0, stream>>>(hist,
                                                                      partial);
  ew_finalize<<<1, 1, 0, stream>>>(partial, out);
}